// DeepFM_50070728737466
// MI455X (gfx1250) — compile-verified
//
#include <hip/hip_runtime.h>
#include <hip/hip_bf16.h>

typedef float v2f __attribute__((ext_vector_type(2)));
typedef float v8f __attribute__((ext_vector_type(8)));

#define BATCH   16384
#define FIELDS  39
#define EMBED   32
#define D_IN    (FIELDS * EMBED)        // 1248
#define HID     400
#define D_CAT   (FIELDS + EMBED + HID)  // 471

// ---------------------------------------------------------------------------
// CDNA5 async global->LDS copy (tracked by ASYNCcnt), CDNA5 ISA.
// VDST = LDS byte address (VGPR), VADDR = 64-bit global address (VGPR pair).
// Generic pointers to LDS carry the LDS byte offset in their low 32 bits.
// ---------------------------------------------------------------------------
__device__ __forceinline__ void async_copy_b128(void* lds_dst, const float* gsrc)
{
    unsigned lds_off = (unsigned)(unsigned long long)lds_dst;
    unsigned long long gaddr = (unsigned long long)gsrc;
    asm volatile("global_load_async_to_lds_b128 %0, %1, off"
                 :: "v"(lds_off), "v"(gaddr)
                 : "memory");
}
__device__ __forceinline__ void wait_async0()
{
    asm volatile("s_wait_asynccnt 0x0" ::: "memory");
}

// ---------------------------------------------------------------------------
// Kernel 1: embedding gather + FM first/second order.
// One wave32 per batch row; lane = embedding dim (each field = one fully
// coalesced 128B embedding-row load).
// ---------------------------------------------------------------------------
__global__ __launch_bounds__(256) void embed_fm_kernel(
    const int*   __restrict__ fidx,
    const float* __restrict__ fval,
    const float* __restrict__ femb,
    const float* __restrict__ fbias,
    float* __restrict__ emb,         // [B, 1248] workspace
    float* __restrict__ out_first,   // [B, 39]
    float* __restrict__ out_second)  // [B, 32]
{
    const int lane = threadIdx.x & 31;
    const int wave = threadIdx.x >> 5;
    const int b    = blockIdx.x * 8 + wave;

    const int   idxA = fidx[(size_t)b * FIELDS + lane];
    const float fvA  = fval[(size_t)b * FIELDS + lane];
    int   idxB = 0;
    float fvB  = 0.0f;
    if (lane < FIELDS - 32) {
        idxB = fidx[(size_t)b * FIELDS + 32 + lane];
        fvB  = fval[(size_t)b * FIELDS + 32 + lane];
    }

    float s = 0.0f, s2 = 0.0f;
    float* embRow = emb + (size_t)b * D_IN;
    #pragma unroll 4
    for (int f = 0; f < FIELDS; ++f) {
        const int   idx = (f < 32) ? __shfl(idxA, f) : __shfl(idxB, f - 32);
        const float fv  = (f < 32) ? __shfl(fvA, f)  : __shfl(fvB, f - 32);
        const float e   = femb[(size_t)idx * EMBED + lane] * fv;
        embRow[f * EMBED + lane] = e;
        s  += e;
        s2 += e * e;
    }
    out_second[(size_t)b * EMBED + lane] = 0.5f * (s * s - s2);

    out_first[(size_t)b * FIELDS + lane] = fbias[idxA] * fvA;
    if (lane < FIELDS - 32)
        out_first[(size_t)b * FIELDS + 32 + lane] = fbias[idxB] * fvB;
}

// ---------------------------------------------------------------------------
// Kernel 2: fp32 GEMM  C = relu(A @ W^T + bias)  via V_WMMA_F32_16X16X4_F32.
// A: [M,K] row-major, W: [N,K] row-major, C: [M,N].
// Block = 8 waves, tile 128(M) x 32(N); each wave owns 16x32 (two 16x16
// accumulators sharing one A fragment). K staged in chunks of 16 through
// double-buffered LDS filled by async global->LDS B128 DMA, overlapped with
// WMMA compute on the other buffer.
//
// LDS row stride = 20 floats (80B): every B128 destination is 16B-aligned
// and the 16 fragment rows land in distinct banks (gcd(20,64)=4 -> 16-row
// cycle; 20*d==2 mod 64 has no solution so lane-halves don't collide).
// ---------------------------------------------------------------------------
__global__ __launch_bounds__(256) void gemm_relu_wmma(
    const float* __restrict__ A,
    const float* __restrict__ W,
    const float* __restrict__ bias,
    float* __restrict__ C,
    int M, int N, int K, int do_relu)
{
    __shared__ float Ash[2][128][20];
    __shared__ float Wsh[2][32][20];

    const int tid  = threadIdx.x;
    const int wave = tid >> 5;
    const int lane = tid & 31;
    const int half = lane >> 4;      // 0: K{0,1}, 1: K{2,3}
    const int r    = lane & 15;

    const int m0 = blockIdx.x * 128;
    const int n0 = blockIdx.y * 32;

    v8f acc0 = {0.f, 0.f, 0.f, 0.f, 0.f, 0.f, 0.f, 0.f};
    v8f acc1 = {0.f, 0.f, 0.f, 0.f, 0.f, 0.f, 0.f, 0.f};

    // issue async DMA for one K-chunk into LDS buffer `buf`
    // A tile: 128 rows x 16 floats = 512 B128 chunks (2 per thread)
    // W tile:  32 rows x 16 floats = 128 B128 chunks (threads 0..127)
    auto issue_chunk = [&](int k0, int buf) {
        #pragma unroll
        for (int i = 0; i < 2; ++i) {
            const int c   = tid + (i << 8);
            const int row = c >> 2;
            const int cb  = (c & 3) << 2;
            async_copy_b128(&Ash[buf][row][cb],
                            A + (size_t)(m0 + row) * K + (k0 + cb));
        }
        if (tid < 128) {
            const int n  = tid >> 2;
            const int cb = (tid & 3) << 2;
            if (n0 + n < N)
                async_copy_b128(&Wsh[buf][n][cb],
                                W + (size_t)(n0 + n) * K + (k0 + cb));
        }
    };

    issue_chunk(0, 0);
    wait_async0();
    __syncthreads();

    int cur = 0;
    for (int k0 = 0; k0 < K; k0 += 16) {
        if (k0 + 16 < K)
            issue_chunk(k0 + 16, cur ^ 1);   // overlaps with compute below

        const float* arow = &Ash[cur][wave * 16 + r][0];
        const float* w0   = &Wsh[cur][r][0];
        const float* w1   = &Wsh[cur][16 + r][0];
        #pragma unroll
        for (int kk = 0; kk < 16; kk += 4) {
            v2f a, b0, b1;
            a.x  = arow[kk + 2 * half];
            a.y  = arow[kk + 2 * half + 1];
            b0.x = w0[kk + 2 * half];
            b0.y = w0[kk + 2 * half + 1];
            b1.x = w1[kk + 2 * half];
            b1.y = w1[kk + 2 * half + 1];
            acc0 = __builtin_amdgcn_wmma_f32_16x16x4_f32(
                false, a, false, b0, (short)0, acc0, false, false);
            acc1 = __builtin_amdgcn_wmma_f32_16x16x4_f32(
                false, a, false, b1, (short)0, acc1, false, false);
        }

        wait_async0();
        __syncthreads();
        cur ^= 1;
    }

    // ---- epilogue: bias + relu, store both 16x16 D tiles ----
    const int   col0 = n0 + r;              // always < N (n0 <= 384, N = 400)
    const int   col1 = n0 + 16 + r;
    const float bn0  = bias[col0];
    const float bn1  = (col1 < N) ? bias[col1] : 0.0f;
    #pragma unroll
    for (int j = 0; j < 8; ++j) {
        const int row = m0 + wave * 16 + j + 8 * half;
        float v0 = acc0[j] + bn0;
        if (do_relu) v0 = fmaxf(v0, 0.0f);
        C[(size_t)row * N + col0] = v0;
        if (col1 < N) {
            float v1 = acc1[j] + bn1;
            if (do_relu) v1 = fmaxf(v1, 0.0f);
            C[(size_t)row * N + col1] = v1;
        }
    }
}

// ---------------------------------------------------------------------------
// Kernel 3: final projection over concat([first(39), second(32), deep(400)]).
// ---------------------------------------------------------------------------
__global__ __launch_bounds__(256) void final_kernel(
    const float* __restrict__ out_first,
    const float* __restrict__ out_second,
    const float* __restrict__ out_deep,
    const float* __restrict__ fW,
    const float* __restrict__ fb,
    float* __restrict__ out0)
{
    const int lane = threadIdx.x & 31;
    const int wave = threadIdx.x >> 5;
    const int b    = blockIdx.x * 8 + wave;

    float acc = 0.0f;
    for (int j = lane; j < D_CAT; j += 32) {
        float c;
        if (j < FIELDS)              c = out_first[(size_t)b * FIELDS + j];
        else if (j < FIELDS + EMBED) c = out_second[(size_t)b * EMBED + (j - FIELDS)];
        else                         c = out_deep[(size_t)b * HID + (j - FIELDS - EMBED)];
        acc += c * fW[j];
    }
    #pragma unroll
    for (int off = 16; off > 0; off >>= 1)
        acc += __shfl_xor(acc, off);
    if (lane == 0)
        out0[b] = acc + fb[0];
}

// ---------------------------------------------------------------------------
extern "C" void kernel_launch(void* const* d_in, const int* in_sizes, int n_in,
                              void* d_out, int out_size, void* d_ws, size_t ws_size,
                              hipStream_t stream)
{
    const int*   fidx  = (const int*)  d_in[0];
    const float* fval  = (const float*)d_in[1];
    const float* femb  = (const float*)d_in[2];
    const float* fbias = (const float*)d_in[3];
    const float* W1    = (const float*)d_in[4];
    const float* b1    = (const float*)d_in[5];
    const float* W2    = (const float*)d_in[6];
    const float* b2    = (const float*)d_in[7];
    const float* fW    = (const float*)d_in[8];
    const float* fb    = (const float*)d_in[9];

    // d_out layout: output(B) | first_order(B*39) | second_order(B*32) | deep_y(B*400)
    float* out0       = (float*)d_out;
    float* out_first  = out0 + BATCH;
    float* out_second = out_first + (size_t)BATCH * FIELDS;
    float* out_deep   = out_second + (size_t)BATCH * EMBED;

    // workspace: emb (B x 1248) | H1 (B x 400)
    float* emb = (float*)d_ws;
    float* h1  = emb + (size_t)BATCH * D_IN;

    embed_fm_kernel<<<BATCH / 8, 256, 0, stream>>>(
        fidx, fval, femb, fbias, emb, out_first, out_second);

    dim3 g1(BATCH / 128, (HID + 31) / 32);   // (128, 13)
    gemm_relu_wmma<<<g1, 256, 0, stream>>>(emb, W1, b1, h1,
                                           BATCH, HID, D_IN, 1);
    gemm_relu_wmma<<<g1, 256, 0, stream>>>(h1, W2, b2, out_deep,
                                           BATCH, HID, HID, 1);

    final_kernel<<<BATCH / 8, 256, 0, stream>>>(
        out_first, out_second, out_deep, fW, fb, out0);
}